// Sampler_13520557047988
// MI455X (gfx1250) — compile-verified
//
#include <hip/hip_runtime.h>
#include <stdint.h>

// Top-k (k=50) temperature sampling, B=256 rows x V=128000 fp32 logits.
// One 1024-thread (32-wave) workgroup per row.
//   Stage A: histogram a 8192-element sample -> conservative value bound below kth
//   Stage B: SINGLE streaming pass, float-compare filter, rare DS appends to LDS
//   Stage C: exact kth key via in-LDS radix select over ~250 candidates,
//            Gumbel-max argmax over survivors (deterministic, seed 42).

#define THREADS 1024
#define CAP     4096   // candidate list capacity (expected ~250 used)
#define SAMPLE  8192   // sample size for bound estimation

// ---- monotonic key: larger float <=> larger uint (no NaNs in input) ----
__device__ __forceinline__ uint32_t order_key(float x) {
    uint32_t u = __float_as_uint(x);
    uint32_t m = (uint32_t)((int32_t)u >> 31);      // all-ones if negative
    return u ^ (m | 0x80000000u);
}
__device__ __forceinline__ float key_to_float(uint32_t key) {
    uint32_t u = (key & 0x80000000u) ? (key ^ 0x80000000u) : ~key;
    return __uint_as_float(u);
}

// ---- deterministic Gumbel(0,1) noise from (seed=42, row, col) ----
__device__ __forceinline__ float gumbel_noise(uint32_t row, uint32_t col) {
    uint64_t x = (42ULL << 48) ^ ((uint64_t)row << 32) ^ (uint64_t)col;
    x += 0x9E3779B97F4A7C15ULL;                      // splitmix64
    x = (x ^ (x >> 30)) * 0xBF58476D1CE4E5B9ULL;
    x = (x ^ (x >> 27)) * 0x94D049BB133111EBULL;
    x ^= (x >> 31);
    float u = ((float)(uint32_t)(x >> 40) + 0.5f) * (1.0f / 16777216.0f); // (0,1)
    float e = -__logf(u);                            // Exp(1) > 0
    return -__logf(e);                               // Gumbel(0,1)
}

struct SelectScratch {
    uint32_t groupSum[64];
    int bucket;
    int krem;
};

// Descending-rank bucket select: highest bucket where cumulative-from-top >= kWanted.
// Parallel 32-bin group sums + short serial scan. All threads call; results in s->.
__device__ void hist_select(const uint32_t* hist, int nbins, int kWanted,
                            SelectScratch* s, int tid) {
    int ngroups = nbins >> 5;
    if (tid < ngroups) {
        uint32_t acc = 0;
        const uint32_t* h = hist + (tid << 5);
#pragma unroll
        for (int b = 0; b < 32; ++b) acc += h[b];
        s->groupSum[tid] = acc;
    }
    __syncthreads();
    if (tid == 0) {
        int cum = 0;
        int g = ngroups - 1;
        for (; g > 0; --g) {
            int c = (int)s->groupSum[g];
            if (cum + c >= kWanted) break;
            cum += c;
        }
        int b = 31;
        for (; b > 0; --b) {
            int c = (int)hist[(g << 5) + b];
            if (cum + c >= kWanted) break;
            cum += c;
        }
        s->bucket = (g << 5) + b;
        s->krem   = kWanted - cum;   // 1-based rank inside chosen bucket
    }
    __syncthreads();
}

extern "C" __global__ __launch_bounds__(THREADS, 1)
void Sampler_13520557047988_kernel(const float* __restrict__ logits,
                                   const float* __restrict__ temps,
                                   const int*   __restrict__ topk_p,
                                   int*         __restrict__ out,
                                   int V) {
    __shared__ uint32_t      hist[2048];
    __shared__ SelectScratch sel;
    __shared__ float         candVal[CAP];
    __shared__ int           candIdx[CAP];
    __shared__ int           candCount;
    __shared__ float         wScore[32];
    __shared__ int           wIdx[32];

    const int row  = blockIdx.x;
    const int tid  = threadIdx.x;
    const int nthr = blockDim.x;

    const float invT = 1.0f / temps[row];
    int k = topk_p[0];
    if (k < 1) k = 1;
    if (k > V) k = V;

    const float*  rowp = logits + (size_t)row * (size_t)V;
    const float4* rp4  = (const float4*)rowp;
    const int     V4   = V >> 2;
    const int     tail = V4 << 2;
    const int     S    = (V < SAMPLE) ? V : SAMPLE;

    // ---------------- Stage A: sample histogram (first S elements) ----------------
    for (int i = tid; i < 2048; i += nthr) hist[i] = 0;
    __syncthreads();
    for (int i = tid; i < S; i += nthr)
        atomicAdd(&hist[order_key(rowp[i] * invT) >> 21], 1u);
    __syncthreads();

    // sample rank: ~5x the expected #samples above the true kth => exp. ~250 survivors
    int r = (int)((5LL * k * S) / V);
    if (r < 4) r = 4;
    if (r > S) r = S;

    // ------------- Stage B: single streaming pass, filter + collect -------------
    int total = 0, nCand = 0;
    for (int attempt = 0; attempt < 4; ++attempt) {
        hist_select(hist, 2048, r, &sel, tid);           // sample hist is preserved
        const uint32_t boundKey = (uint32_t)sel.bucket << 21;  // bucket lower bound
        const float    boundVal = key_to_float(boundKey);      // <= true kth w.h.p.
        if (tid == 0) candCount = 0;
        __syncthreads();

        for (int i = tid; i < V4; i += nthr) {
            float4 v = rp4[i];
            __builtin_prefetch((const void*)(rp4 + i + nthr), 0, 0); // global_prefetch_b8
            float s0 = v.x * invT, s1 = v.y * invT;
            float s2 = v.z * invT, s3 = v.w * invT;
            if (s0 >= boundVal) { int p = atomicAdd(&candCount, 1); if (p < CAP) { candVal[p] = s0; candIdx[p] = (i << 2) + 0; } }
            if (s1 >= boundVal) { int p = atomicAdd(&candCount, 1); if (p < CAP) { candVal[p] = s1; candIdx[p] = (i << 2) + 1; } }
            if (s2 >= boundVal) { int p = atomicAdd(&candCount, 1); if (p < CAP) { candVal[p] = s2; candIdx[p] = (i << 2) + 2; } }
            if (s3 >= boundVal) { int p = atomicAdd(&candCount, 1); if (p < CAP) { candVal[p] = s3; candIdx[p] = (i << 2) + 3; } }
        }
        for (int j = tail + tid; j < V; j += nthr) {
            float s = rowp[j] * invT;
            if (s >= boundVal) { int p = atomicAdd(&candCount, 1); if (p < CAP) { candVal[p] = s; candIdx[p] = j; } }
        }
        __syncthreads();
        total = candCount;
        nCand = (total > CAP) ? CAP : total;
        if (total <= CAP && nCand >= k) break;           // success (the normal case)
        // retry with adjusted bound (deterministic, same on all threads)
        if (total > CAP) r = (r >> 2) < 1 ? 1 : (r >> 2);   // bound too low  -> raise
        else             r = (r << 3) > S ? S : (r << 3);   // bound too high -> lower
        __syncthreads();
    }
    int kEff = k;
    if (kEff > nCand) kEff = (nCand > 0) ? nCand : 1;    // pathological fallback

    // ------- Stage C: exact kth key among candidates (in-LDS radix 11/11/10) -------
    for (int i = tid; i < 2048; i += nthr) hist[i] = 0;
    __syncthreads();
    for (int i = tid; i < nCand; i += nthr)
        atomicAdd(&hist[order_key(candVal[i]) >> 21], 1u);
    __syncthreads();
    hist_select(hist, 2048, kEff, &sel, tid);
    const uint32_t c1 = (uint32_t)sel.bucket;
    const int      k1 = sel.krem;

    for (int i = tid; i < 2048; i += nthr) hist[i] = 0;
    __syncthreads();
    for (int i = tid; i < nCand; i += nthr) {
        uint32_t key = order_key(candVal[i]);
        if ((key >> 21) == c1) atomicAdd(&hist[(key >> 10) & 0x7FFu], 1u);
    }
    __syncthreads();
    hist_select(hist, 2048, k1, &sel, tid);
    const uint32_t prefix22 = (c1 << 21) | ((uint32_t)sel.bucket << 10);
    const int      k2       = sel.krem;

    for (int i = tid; i < 1024; i += nthr) hist[i] = 0;
    __syncthreads();
    const uint32_t pfx_hi = prefix22 >> 10;
    for (int i = tid; i < nCand; i += nthr) {
        uint32_t key = order_key(candVal[i]);
        if ((key >> 10) == pfx_hi) atomicAdd(&hist[key & 0x3FFu], 1u);
    }
    __syncthreads();
    hist_select(hist, 1024, k2, &sel, tid);
    const uint32_t kthKey = prefix22 | (uint32_t)sel.bucket; // ties at kth are kept

    // -------- Gumbel-max over surviving candidates (key >= kthKey) --------
    float best    = -3.402823466e38f;
    int   bestIdx = 0x7FFFFFFF;
    for (int i = tid; i < nCand; i += nthr) {
        float s = candVal[i];
        if (order_key(s) >= kthKey) {
            int   j  = candIdx[i];
            float sc = s + gumbel_noise((uint32_t)row, (uint32_t)j);
            if (sc > best || (sc == best && j < bestIdx)) { best = sc; bestIdx = j; }
        }
    }
    // wave32 argmax reduce
#pragma unroll
    for (int off = 16; off > 0; off >>= 1) {
        float os = __shfl_xor(best, off, 32);
        int   oi = __shfl_xor(bestIdx, off, 32);
        if (os > best || (os == best && oi < bestIdx)) { best = os; bestIdx = oi; }
    }
    int wid = tid >> 5, lane = tid & 31;
    if (lane == 0) { wScore[wid] = best; wIdx[wid] = bestIdx; }
    __syncthreads();
    if (tid < 32) {
        int nw = nthr >> 5;
        best    = (tid < nw) ? wScore[tid] : -3.402823466e38f;
        bestIdx = (tid < nw) ? wIdx[tid]   : 0x7FFFFFFF;
#pragma unroll
        for (int off = 16; off > 0; off >>= 1) {
            float os = __shfl_xor(best, off, 32);
            int   oi = __shfl_xor(bestIdx, off, 32);
            if (os > best || (os == best && oi < bestIdx)) { best = os; bestIdx = oi; }
        }
        if (tid == 0) out[row] = bestIdx;
    }
}

extern "C" void kernel_launch(void* const* d_in, const int* in_sizes, int n_in,
                              void* d_out, int out_size, void* d_ws, size_t ws_size,
                              hipStream_t stream) {
    (void)n_in; (void)d_ws; (void)ws_size; (void)out_size;
    const float* logits = (const float*)d_in[0];
    const float* temps  = (const float*)d_in[1];
    const int*   topk   = (const int*)d_in[2];
    const int B = in_sizes[1];                 // temperatures: one per row
    const int V = in_sizes[0] / B;             // 128000
    int* out = (int*)d_out;                    // token indices (int32)
    hipLaunchKernelGGL(Sampler_13520557047988_kernel,
                       dim3(B), dim3(THREADS), 0, stream,
                       logits, temps, topk, out, V);
}